// TextLevelGNN_9337258901945
// MI455X (gfx1250) — compile-verified
//
#include <hip/hip_runtime.h>
#include <hip/hip_bf16.h>
#include <math.h>

#define LN_EPS 1e-5f

typedef __attribute__((ext_vector_type(2))) float v2f;
typedef __attribute__((ext_vector_type(8))) float v8f;

static constexpr int B_  = 64;
static constexpr int L_  = 512;
static constexpr int D2_ = 16;   // 2 * n_degree
static constexpr int D_  = 256;  // d_model
static constexpr int C_  = 4;    // n_category

static constexpr int POS_PER_WAVE   = 8;
static constexpr int WAVES_PER_BLK  = 8;
static constexpr int POS_PER_BLOCK  = POS_PER_WAVE * WAVES_PER_BLK; // 64

// ---------------------------------------------------------------------------
// Kernel 0: zero the (B, D) accumulator in workspace
// ---------------------------------------------------------------------------
__global__ void zero_ws_kernel(float* __restrict__ ws, int n) {
    int i = blockIdx.x * blockDim.x + threadIdx.x;
    if (i < n) ws[i] = 0.0f;
}

// ---------------------------------------------------------------------------
// Kernel 1: gather + layernorm + weighted-neighbor-max + eta blend,
// accumulated over L into H (B, D).  One wave per 8 consecutive (b,l) positions.
// Lane c-mapping: channels lane*4+i (i<4) and 128+lane*4+(i-4) (i>=4).
// ---------------------------------------------------------------------------
__global__ __launch_bounds__(256) void gnn_node_kernel(
    const int*   __restrict__ x,            // (B, L)
    const int*   __restrict__ nb_x,         // (B, L, D2)
    const int*   __restrict__ w_edge,       // (B, L, D2)
    const float* __restrict__ emb,          // (V, D)
    const float* __restrict__ w_edge_table, // (E, 1)
    const float* __restrict__ eta_table,    // (V, 1)
    const float* __restrict__ ln_gamma,     // (D,)
    const float* __restrict__ ln_beta,      // (D,)
    float*       __restrict__ H)            // (B, D) accumulator
{
    __shared__ float lds_h[D_];

    const int lane = threadIdx.x & 31;
    const int wave = threadIdx.x >> 5;
    const int b    = blockIdx.y;
    const int l0   = blockIdx.x * POS_PER_BLOCK + wave * POS_PER_WAVE;

    for (int c = threadIdx.x; c < D_; c += blockDim.x) lds_h[c] = 0.0f;
    __syncthreads();

    // per-lane gamma/beta (coalesced b128 loads, hot in L0 after first block)
    float g[8], bt[8];
    {
        float4 q0 = *(const float4*)(ln_gamma + lane * 4);
        float4 q1 = *(const float4*)(ln_gamma + 128 + lane * 4);
        g[0]=q0.x; g[1]=q0.y; g[2]=q0.z; g[3]=q0.w;
        g[4]=q1.x; g[5]=q1.y; g[6]=q1.z; g[7]=q1.w;
        float4 p0 = *(const float4*)(ln_beta + lane * 4);
        float4 p1 = *(const float4*)(ln_beta + 128 + lane * 4);
        bt[0]=p0.x; bt[1]=p0.y; bt[2]=p0.z; bt[3]=p0.w;
        bt[4]=p1.x; bt[5]=p1.y; bt[6]=p1.z; bt[7]=p1.w;
    }

    float acc[8];
    #pragma unroll
    for (int i = 0; i < 8; ++i) acc[i] = 0.0f;

    for (int p = 0; p < POS_PER_WAVE; ++p) {
        const int l   = l0 + p;
        const int pos = b * L_ + l;
        const int self_idx = x[pos];

        // prefetch the self row; it is consumed last (node == D2)
        __builtin_prefetch(emb + (size_t)self_idx * D_ + lane * 4, 0, 3);

        // lanes 0..15: neighbor index and pre-gathered edge weight.
        // Clamp address for lanes >= D2 (value unused) to keep loads
        // unconditional (no exec branching).
        const int j = lane & (D2_ - 1);
        int   nb_idx = nb_x[pos * D2_ + j];
        float wj     = w_edge_table[w_edge[pos * D2_ + j]];

        float msg[8], selfn[8];
        #pragma unroll
        for (int i = 0; i < 8; ++i) { msg[i] = -INFINITY; selfn[i] = 0.0f; }

        for (int node = 0; node <= D2_; ++node) {
            const int idx = (node < D2_) ? __shfl(nb_idx, node) : self_idx;
            const float* row = emb + (size_t)idx * D_;

            float e[8];
            {
                float4 q0 = *(const float4*)(row + lane * 4);        // b128
                float4 q1 = *(const float4*)(row + 128 + lane * 4);  // b128
                e[0]=q0.x; e[1]=q0.y; e[2]=q0.z; e[3]=q0.w;
                e[4]=q1.x; e[5]=q1.y; e[6]=q1.z; e[7]=q1.w;
            }

            // mean over 256 (8 per lane + 5-step wave32 butterfly)
            float s = 0.0f;
            #pragma unroll
            for (int i = 0; i < 8; ++i) s += e[i];
            #pragma unroll
            for (int m = 16; m >= 1; m >>= 1) s += __shfl_xor(s, m);
            const float mu = s * (1.0f / 256.0f);

            float d[8], v = 0.0f;
            #pragma unroll
            for (int i = 0; i < 8; ++i) { d[i] = e[i] - mu; v += d[i] * d[i]; }
            #pragma unroll
            for (int m = 16; m >= 1; m >>= 1) v += __shfl_xor(v, m);
            const float inv = rsqrtf(v * (1.0f / 256.0f) + LN_EPS);

            float n[8];
            #pragma unroll
            for (int i = 0; i < 8; ++i) n[i] = d[i] * inv * g[i] + bt[i];

            if (node < D2_) {
                const float w = __shfl(wj, node);
                #pragma unroll
                for (int i = 0; i < 8; ++i) msg[i] = fmaxf(msg[i], w * n[i]);
            } else {
                #pragma unroll
                for (int i = 0; i < 8; ++i) selfn[i] = n[i];
            }
        }

        const float eta = eta_table[self_idx];
        const float om  = 1.0f - eta;
        #pragma unroll
        for (int i = 0; i < 8; ++i) acc[i] += om * msg[i] + eta * selfn[i];
    }

    // cross-wave reduce via LDS float atomics (ds_add_f32)
    #pragma unroll
    for (int i = 0; i < 8; ++i) {
        const int ch = (i < 4) ? (lane * 4 + i) : (128 + lane * 4 + (i - 4));
        atomicAdd(&lds_h[ch], acc[i]);
    }
    __syncthreads();

    if (threadIdx.x < D_)
        atomicAdd(&H[b * D_ + threadIdx.x], lds_h[threadIdx.x]);
}

// ---------------------------------------------------------------------------
// Kernel 2: scores = H (64,256) @ fc_w (256,4) + fc_b  via V_WMMA_F32_16X16X4_F32
// 4 waves; wave w owns batch rows [16w, 16w+16). N padded 4 -> 16.
//
// B (256x16, zero-padded) is staged once in LDS with a pair-interleaved
// layout so each lane fetches {B[ka][n], B[ka+1][n]} as one ds_load_b64:
//   Bl[((k>>1)*16 + n)*2 + (k&1)] = B[k][n]        (ka is always even)
//
// A frag: lanes 0-15 M=lane K={k0,k0+1}; lanes 16-31 M=lane-16 K={k0+2,k0+3}.
// D:      vgpr r: lanes 0-15 -> M=r, N=lane; lanes 16-31 -> M=8+r, N=lane-16.
// Two interleaved accumulators break the WMMA D->C dependency chain.
// ---------------------------------------------------------------------------
__global__ __launch_bounds__(128) void score_wmma_kernel(
    const float* __restrict__ H,     // (B, D)
    const float* __restrict__ fc_w,  // (D, C)
    const float* __restrict__ fc_b,  // (C,)
    float*       __restrict__ out)   // (B, C)
{
    __shared__ float Bl[D_ * 16];    // 16 KB, pair-interleaved padded B

    // Fill: unconditional (clamped) loads + select; no exec branching.
    for (int i = threadIdx.x; i < D_ * 16; i += blockDim.x) {
        const int k = i >> 4;
        const int n = i & 15;
        const float t = fc_w[k * C_ + (n & (C_ - 1))];   // always in-bounds
        Bl[((k >> 1) * 16 + n) * 2 + (k & 1)] = (n < C_) ? t : 0.0f;
    }
    __syncthreads();

    const int lane = threadIdx.x & 31;
    const int wave = threadIdx.x >> 5;   // 0..3
    const int row0 = wave * 16;
    const int half = lane >> 4;          // 0 or 1 (selects K pair)
    const int lm   = lane & 15;          // M for A-frag, N for B-frag

    v8f acc0 = {0.f,0.f,0.f,0.f,0.f,0.f,0.f,0.f};
    v8f acc1 = {0.f,0.f,0.f,0.f,0.f,0.f,0.f,0.f};
    const float* arow = H + (row0 + lm) * D_;

    for (int k0 = 0; k0 < D_; k0 += 8) {
        const int ka = k0 + 2 * half;        // even
        const int kb = k0 + 4 + 2 * half;    // even

        v2f a0; a0.x = arow[ka];     a0.y = arow[ka + 1];
        v2f a1; a1.x = arow[kb];     a1.y = arow[kb + 1];
        v2f b0 = *(const v2f*)&Bl[((ka >> 1) * 16 + lm) * 2];   // ds_load_b64
        v2f b1 = *(const v2f*)&Bl[((kb >> 1) * 16 + lm) * 2];   // ds_load_b64

        acc0 = __builtin_amdgcn_wmma_f32_16x16x4_f32(
            false, a0, false, b0, (short)0, acc0, false, false);
        acc1 = __builtin_amdgcn_wmma_f32_16x16x4_f32(
            false, a1, false, b1, (short)0, acc1, false, false);
    }

    const v8f acc = acc0 + acc1;

    if (lm < C_) {
        const float bias  = fc_b[lm];
        const int   mbase = row0 + 8 * half;
        #pragma unroll
        for (int r = 0; r < 8; ++r)
            out[(mbase + r) * C_ + lm] = acc[r] + bias;
    }
}

// ---------------------------------------------------------------------------
extern "C" void kernel_launch(void* const* d_in, const int* in_sizes, int n_in,
                              void* d_out, int out_size, void* d_ws, size_t ws_size,
                              hipStream_t stream) {
    const int*   x            = (const int*)  d_in[0];
    const int*   nb_x         = (const int*)  d_in[1];
    const int*   w_edge       = (const int*)  d_in[2];
    const float* emb_weight   = (const float*)d_in[3];
    const float* w_edge_table = (const float*)d_in[4];
    const float* eta_table    = (const float*)d_in[5];
    const float* ln_gamma     = (const float*)d_in[6];
    const float* ln_beta      = (const float*)d_in[7];
    const float* fc_w         = (const float*)d_in[8];
    const float* fc_b         = (const float*)d_in[9];
    float*       out          = (float*)d_out;
    float*       H            = (float*)d_ws;      // (B, D) = 64 KB

    // 0) zero accumulator
    zero_ws_kernel<<<(B_ * D_ + 255) / 256, 256, 0, stream>>>(H, B_ * D_);

    // 1) gather / layernorm / max / blend, summed over L into H
    dim3 grid1(L_ / POS_PER_BLOCK, B_);   // (8, 64)
    gnn_node_kernel<<<grid1, 256, 0, stream>>>(
        x, nb_x, w_edge, emb_weight, w_edge_table, eta_table,
        ln_gamma, ln_beta, H);

    // 2) WMMA epilogue: scores = H @ fc_w + fc_b
    score_wmma_kernel<<<1, 128, 0, stream>>>(H, fc_w, fc_b, out);
}